// DiscreteKalmanFilter_33414845563530
// MI455X (gfx1250) — compile-verified
//
#include <hip/hip_runtime.h>
#include <math.h>

typedef float v2f __attribute__((ext_vector_type(2)));
typedef float v8f __attribute__((ext_vector_type(8)));

#define T_STEPS 4096
#define DIM     64
#define LATENT  128
#define LDL     129   // padded ld for 128-wide matrices (conflict-free A-side reads)
#define LDD     65    // padded ld for 64-wide matrices
#define NTHREADS 512  // 16 waves (wave32) on one WGP
#define CONV_EPS 1e-6f

// dst(MxN) = A(MxK) @ (transB ? B^T : B) [+ dst if accumulate], f32 WMMA 16x16x4.
__device__ __forceinline__ void mm_wmma(
    float* dst, int ldd_,
    const float* A, int lda,
    const float* B, int ldb,
    int M, int N, int K, bool transB, bool accumulate)
{
  const int lane  = threadIdx.x & 31;
  const int wave  = threadIdx.x >> 5;
  const int nwav  = NTHREADS >> 5;
  const int half  = lane >> 4;
  const int idx   = lane & 15;
  const int tilesN = N >> 4;
  const int ntiles = (M >> 4) * tilesN;
  for (int t = wave; t < ntiles; t += nwav) {
    const int ti = t / tilesN;
    const int bi = ti << 4;
    const int bj = (t - ti * tilesN) << 4;
    v8f acc = {};
    if (accumulate) {
#pragma unroll
      for (int r = 0; r < 8; ++r) acc[r] = dst[(bi + r + 8 * half) * ldd_ + bj + idx];
    }
    for (int k = 0; k < K; k += 4) {
      const int kk = k + 2 * half;
      v2f a, b;
      a.x = A[(bi + idx) * lda + kk];
      a.y = A[(bi + idx) * lda + kk + 1];
      if (transB) {
        b.x = B[(bj + idx) * ldb + kk];
        b.y = B[(bj + idx) * ldb + kk + 1];
      } else {
        b.x = B[kk * ldb + bj + idx];
        b.y = B[(kk + 1) * ldb + bj + idx];
      }
      acc = __builtin_amdgcn_wmma_f32_16x16x4_f32(false, a, false, b, (short)0, acc,
                                                  false, false);
    }
#pragma unroll
    for (int r = 0; r < 8; ++r) dst[(bi + r + 8 * half) * ldd_ + bj + idx] = acc[r];
  }
}

__global__ __launch_bounds__(NTHREADS, 1)
void kalman_kernel(const float* __restrict__ gz, const float* __restrict__ gF,
                   const float* __restrict__ gH, const float* __restrict__ gQ,
                   const float* __restrict__ gR, float* __restrict__ out,
                   float* __restrict__ wsHP /* 64*128 floats or nullptr */)
{
  __shared__ float sF [LATENT * LDL];   // F
  __shared__ float sT [LATENT * LDL];   // temp: F@P / M=K@HP / KS
  __shared__ float sP [LATENT * LDL];   // P (posterior)
  __shared__ float sH [DIM * LDL];      // H
  __shared__ float sHP[DIM * LDL];      // H@P
  __shared__ float sS [DIM * LDD];      // S -> chol(S) -> restored S (L kept once converged)
  __shared__ float sK [LATENT * LDD];   // Kalman gain
  __shared__ float sx[LATENT], sxt[LATENT], sv[DIM], sSd[DIM];
  __shared__ int s_changed, s_conv;

  const int tid = threadIdx.x;

  for (int i = tid; i < LATENT * LATENT; i += NTHREADS) {
    int r = i >> 7, c = i & 127;
    sF[r * LDL + c] = gF[i];
    sP[r * LDL + c] = (r == c) ? 1.0f : 0.0f;      // P0 = I
  }
  for (int i = tid; i < DIM * LATENT; i += NTHREADS) {
    int r = i >> 7, c = i & 127;
    sH[r * LDL + c] = gH[i];
  }
  if (tid < LATENT) sx[tid] = 0.0f;                // x0 = 0
  if (tid == 0) { s_conv = 0; s_changed = 1; }
  __syncthreads();

  float logp_sum = 0.0f;
  const float LOG2PI = 1.8378770664093453f;

  for (int t = 0; t < T_STEPS; ++t) {
    const bool conv = (s_conv != 0);               // uniform; set behind barriers
    if (tid == 0 && t + 1 < T_STEPS) __builtin_prefetch(&gz[(t + 1) * DIM], 0, 0);

    if (t > 0) {                                   // x = F x (always)
      if (tid < LATENT) {
        float s = 0.0f;
        for (int j = 0; j < LATENT; ++j) s += sF[tid * LDL + j] * sx[j];
        sxt[tid] = s;
      }
      __syncthreads();
      if (tid < LATENT) sx[tid] = sxt[tid];
      __syncthreads();
    }

    if (!conv) {
      // ---------------- full covariance recursion ----------------
      if (t > 0) {
        mm_wmma(sT, LDL, sF, LDL, sP, LDL, LATENT, LATENT, LATENT, false, false); // T=F P
        __syncthreads();
        mm_wmma(sP, LDL, sT, LDL, sF, LDL, LATENT, LATENT, LATENT, true,  false); // P=T F^T
        __syncthreads();
        if (tid < LATENT) { float q = gQ[tid]; sP[tid * LDL + tid] += q * q; }
        __syncthreads();
      }
      mm_wmma(sHP, LDL, sH, LDL, sP, LDL, DIM, LATENT, LATENT, false, false);     // HP=H P
      __syncthreads();

      // steady-state detection: HP determines S, K, L, logdet
      if (tid == 0) s_changed = (t == 0 || wsHP == nullptr) ? 1 : 0;
      __syncthreads();
      if (wsHP != nullptr) {
        for (int i2 = tid; i2 < DIM * LATENT; i2 += NTHREADS) {
          int r = i2 >> 7, c = i2 & 127;
          float cur = sHP[r * LDL + c];
          if (t > 0 && fabsf(cur - wsHP[i2]) > CONV_EPS) s_changed = 1;
          wsHP[i2] = cur;
        }
      }
      __syncthreads();
      const bool nowconv = (s_changed == 0);
      if (tid == 0 && nowconv) s_conv = 1;

      mm_wmma(sS, LDD, sHP, LDL, sH, LDL, DIM, DIM, LATENT, true, false);         // S=HP H^T
      __syncthreads();
      if (tid < DIM) {
        float r = gR[tid];
        float d = sS[tid * LDD + tid] + r * r;                                    // +diag(R)
        sS[tid * LDD + tid] = d;
        sSd[tid] = d;
        float s = 0.0f;                                                           // v=z-Hx
        for (int j = 0; j < LATENT; ++j) s += sH[tid * LDL + j] * sx[j];
        sv[tid] = gz[t * DIM + tid] - s;
      }
      __syncthreads();

      if (tid < 32) {                              // chol(S) lower, wave-0 lockstep
        volatile float* S = sS;
        const int lane = tid;
        for (int j = 0; j < DIM; ++j) {
          float d = sqrtf(S[j * LDD + j]);
          float inv = 1.0f / d;
          if (lane == 0) S[j * LDD + j] = d;
          for (int i = j + 1 + lane; i < DIM; i += 32) S[i * LDD + j] *= inv;
          for (int kc = j + 1; kc < DIM; ++kc) {
            float ljk = S[kc * LDD + j];
            for (int i = kc + lane; i < DIM; i += 32) S[i * LDD + kc] -= S[i * LDD + j] * ljk;
          }
        }
      }
      __syncthreads();

      if (!nowconv && tid < LATENT) {              // K row r: S^{-1} HP[:,r] in place
        const int r = tid;
        float* krow = &sK[r * LDD];
        for (int j = 0; j < DIM; ++j) {
          float s = sHP[j * LDL + r];
          for (int k2 = 0; k2 < j; ++k2) s -= sS[j * LDD + k2] * krow[k2];
          krow[j] = s / sS[j * LDD + j];
        }
        for (int j = DIM - 1; j >= 0; --j) {
          float s = krow[j];
          for (int k2 = j + 1; k2 < DIM; ++k2) s -= sS[k2 * LDD + j] * krow[k2];
          krow[j] = s / sS[j * LDD + j];
        }
      }
      if (tid >= 128 && tid < 160) {               // NLL: y=L^{-1}v column sweep, wave 4
        const int lane = tid - 128;
        float a0 = sv[lane], a1 = sv[lane + 32];
        float quad = 0.0f;
        float ld = logf(sS[lane * LDD + lane]) + logf(sS[(lane + 32) * LDD + lane + 32]);
        for (int off = 16; off; off >>= 1) ld += __shfl_xor(ld, off, 32);
        for (int j = 0; j < DIM; ++j) {
          float Ljj = sS[j * LDD + j];
          float src = (j < 32) ? a0 : a1;
          float yj = __shfl(src, j & 31, 32) / Ljj;
          float l0 = sS[lane * LDD + j];
          float l1 = sS[(lane + 32) * LDD + j];
          if (lane > j)      a0 -= l0 * yj;
          if (lane + 32 > j) a1 -= l1 * yj;
          quad += yj * yj;
        }
        if (lane == 0) logp_sum += -0.5f * ((float)DIM * LOG2PI + 2.0f * ld + quad);
      }
      __syncthreads();

      if (!nowconv) {                              // restore S (lower from upper, diag saved)
        for (int i2 = tid; i2 < DIM * DIM; i2 += NTHREADS) {
          int ri = i2 >> 6, ci = i2 & 63;
          if (ri > ci)       sS[ri * LDD + ci] = sS[ci * LDD + ri];
          else if (ri == ci) sS[ri * LDD + ci] = sSd[ri];
        }
      }
      if (tid < LATENT) {                          // x += K v
        float s = 0.0f;
        for (int j = 0; j < DIM; ++j) s += sK[tid * LDD + j] * sv[j];
        sx[tid] += s;
      }
      if (!nowconv)
        mm_wmma(sT, LDL, sK, LDD, sHP, LDL, LATENT, LATENT, DIM, false, false);   // M=K HP
      __syncthreads();
      if (!nowconv) {
        for (int i2 = tid; i2 < LATENT * LATENT; i2 += NTHREADS) {                // P-=M+M^T
          int ri = i2 >> 7, ci = i2 & 127;
          sP[ri * LDL + ci] -= sT[ri * LDL + ci] + sT[ci * LDL + ri];
        }
        __syncthreads();
        mm_wmma(sT, LDD, sK, LDD, sS, LDD, LATENT, DIM, DIM, false, false);       // KS=K S
        __syncthreads();
        mm_wmma(sP, LDL, sT, LDD, sK, LDD, LATENT, LATENT, DIM, true, true);      // P+=KS K^T
        __syncthreads();
      }
    } else {
      // ---------------- steady state: covariance frozen (sS holds L, sK holds K) ----------
      if (tid < DIM) {
        float s = 0.0f;
        for (int j = 0; j < LATENT; ++j) s += sH[tid * LDL + j] * sx[j];
        sv[tid] = gz[t * DIM + tid] - s;
      }
      __syncthreads();
      if (tid >= 128 && tid < 160) {
        const int lane = tid - 128;
        float a0 = sv[lane], a1 = sv[lane + 32];
        float quad = 0.0f;
        float ld = logf(sS[lane * LDD + lane]) + logf(sS[(lane + 32) * LDD + lane + 32]);
        for (int off = 16; off; off >>= 1) ld += __shfl_xor(ld, off, 32);
        for (int j = 0; j < DIM; ++j) {
          float Ljj = sS[j * LDD + j];
          float src = (j < 32) ? a0 : a1;
          float yj = __shfl(src, j & 31, 32) / Ljj;
          float l0 = sS[lane * LDD + j];
          float l1 = sS[(lane + 32) * LDD + j];
          if (lane > j)      a0 -= l0 * yj;
          if (lane + 32 > j) a1 -= l1 * yj;
          quad += yj * yj;
        }
        if (lane == 0) logp_sum += -0.5f * ((float)DIM * LOG2PI + 2.0f * ld + quad);
      }
      __syncthreads();
      if (tid < LATENT) {
        float s = 0.0f;
        for (int j = 0; j < DIM; ++j) s += sK[tid * LDD + j] * sv[j];
        sx[tid] += s;
      }
      __syncthreads();
    }
  }

  if (tid == 128) out[0] = -(logp_sum / (float)T_STEPS) / (float)DIM;
}

extern "C" void kernel_launch(void* const* d_in, const int* in_sizes, int n_in,
                              void* d_out, int out_size, void* d_ws, size_t ws_size,
                              hipStream_t stream) {
  (void)in_sizes; (void)n_in; (void)out_size;
  const float* z  = (const float*)d_in[0];
  const float* F  = (const float*)d_in[1];
  const float* H  = (const float*)d_in[2];
  const float* pQ = (const float*)d_in[3];
  const float* pR = (const float*)d_in[4];
  float* out = (float*)d_out;
  float* wsHP = (ws_size >= (size_t)(DIM * LATENT) * sizeof(float)) ? (float*)d_ws : nullptr;
  hipLaunchKernelGGL(kalman_kernel, dim3(1), dim3(NTHREADS), 0, stream,
                     z, F, H, pQ, pR, out, wsHP);
}